// LPALM_48576080117916
// MI455X (gfx1250) — compile-verified
//
#include <hip/hip_runtime.h>

#define B_ 32
#define M_ 156
#define N_ 3
#define P_ 9025
#define T_ 25
#define TILES 565          // ceil(9025/16)
#define WPB 4              // waves per block
#define TPW 8              // pixel tiles per wave
#define NWAVES 72          // waves per batch (18 blocks * 4)
#define XSTRIDE 17         // LDS row stride (dwords), bank-conflict free

typedef __attribute__((ext_vector_type(2))) float v2f;
typedef __attribute__((ext_vector_type(8))) float v8f;

// ---------------------------------------------------------------------------
// Kernel 1: S update + per-wave partial C = S S^T (3x3), D = X S^T (156x3)
// u = W^T X - M3 * S_prev ;  S_new = relu(S_prev + u - theta)
// ---------------------------------------------------------------------------
__global__ __launch_bounds__(128) void lpalm_s_update(
    const float* __restrict__ X, const float* __restrict__ Sprev,
    const float* __restrict__ Wall, const float* __restrict__ theta,
    const float* __restrict__ M3ws,
    float* __restrict__ Sout, float* __restrict__ SpredOut,
    float* __restrict__ Cpart, float* __restrict__ Dpart, int t)
{
    __shared__ float Wlds[480];
    __shared__ float Xlds[WPB][160 * XSTRIDE];
    __shared__ float snLds[WPB][3][16];

    const int tid  = threadIdx.x;
    const int lane = tid & 31;
    const int wv   = tid >> 5;
    const int b    = blockIdx.y;
    const int wave = blockIdx.x * WPB + wv;

    // stage W_t (156x3) into LDS
    const float* Wt = Wall + (size_t)t * (M_ * N_);
    for (int i = tid; i < M_ * N_; i += 128) Wlds[i] = Wt[i];
    __syncthreads();

    const float th = theta[t];
    float m3[9];
#pragma unroll
    for (int i = 0; i < 9; ++i) m3[i] = M3ws[b * 16 + i];

    const int c16 = lane & 15;        // pixel-in-tile / row-in-tile index
    const int k0  = (lane >> 4) * 2;  // K offset per WMMA f32 16x16x4 layout
    float* xw = &Xlds[wv][0];

    // zero pad rows 156..159 (phase 2 reads them; phase 1 never writes them)
    xw[(156 + (lane >> 4)) * XSTRIDE + c16] = 0.0f;
    xw[(158 + (lane >> 4)) * XSTRIDE + c16] = 0.0f;

    v8f dacc[10];
#pragma unroll
    for (int mc = 0; mc < 10; ++mc) dacc[mc] = (v8f){0,0,0,0,0,0,0,0};
    float q00=0.f,q01=0.f,q02=0.f,q11=0.f,q12=0.f,q22=0.f;

    const size_t Xb = (size_t)b * M_ * P_;
    const size_t Sb = (size_t)b * N_ * P_;

    for (int it = 0; it < TPW; ++it) {
        const int tile = wave * TPW + it;        // wave-uniform
        if (tile >= TILES) break;                // wave-uniform branch
        const int p0 = tile * 16;
        const int px = p0 + c16;
        const bool pv = (px < P_);

        // ---- phase 1: u(16n x 16p) = W^T X, K over M=156 (39 WMMAs) ----
        // software-pipelined: prefetch next kc's W (LDS) and X (global)
        // fragments into fresh registers before the current WMMA consumes.
        v8f uacc = (v8f){0,0,0,0,0,0,0,0};
        float ax = 0.f, ay = 0.f, bx = 0.f, by = 0.f;
        if (c16 < 3) {
            ax = Wlds[k0 * 3 + c16];
            ay = Wlds[(k0 + 1) * 3 + c16];
        }
        if (pv) {
            bx = X[Xb + (size_t)k0 * P_ + px];
            by = X[Xb + (size_t)(k0 + 1) * P_ + px];
        }
#pragma unroll
        for (int kc = 0; kc < 39; ++kc) {
            const int m0 = kc * 4;
            float nax = 0.f, nay = 0.f, nbx = 0.f, nby = 0.f;
            if (kc < 38) {
                const int m1 = m0 + 4;
                if (c16 < 3) {
                    nax = Wlds[(m1 + k0) * 3 + c16];
                    nay = Wlds[(m1 + k0 + 1) * 3 + c16];
                }
                if (pv) {
                    nbx = X[Xb + (size_t)(m1 + k0) * P_ + px];
                    nby = X[Xb + (size_t)(m1 + k0 + 1) * P_ + px];
                }
            }
            // stage current X data into LDS for phase 2 (same-wave DS order)
            xw[(m0 + k0)     * XSTRIDE + c16] = bx;
            xw[(m0 + k0 + 1) * XSTRIDE + c16] = by;
            v2f af; af.x = ax; af.y = ay;
            v2f bf; bf.x = bx; bf.y = by;
            uacc = __builtin_amdgcn_wmma_f32_16x16x4_f32(
                false, af, false, bf, (short)0, uacc, false, false);
            ax = nax; ay = nay; bx = nbx; by = nby;
        }

        // ---- S update (lanes 0..15 hold u rows n=0..2 in vgprs 0..2) ----
        float s0=0.f, s1=0.f, s2=0.f;
        if (pv) {
            s0 = Sprev[Sb + 0*(size_t)P_ + px];
            s1 = Sprev[Sb + 1*(size_t)P_ + px];
            s2 = Sprev[Sb + 2*(size_t)P_ + px];
        }
        float sn0=0.f, sn1=0.f, sn2=0.f;
        if (lane < 16) {
            const float v0 = m3[0]*s0 + m3[1]*s1 + m3[2]*s2;
            const float v1 = m3[3]*s0 + m3[4]*s1 + m3[5]*s2;
            const float v2 = m3[6]*s0 + m3[7]*s1 + m3[8]*s2;
            sn0 = fmaxf(s0 + uacc[0] - v0 - th, 0.f);
            sn1 = fmaxf(s1 + uacc[1] - v1 - th, 0.f);
            sn2 = fmaxf(s2 + uacc[2] - v2 - th, 0.f);
            if (!pv) { sn0 = 0.f; sn1 = 0.f; sn2 = 0.f; }
            snLds[wv][0][c16] = sn0;
            snLds[wv][1][c16] = sn1;
            snLds[wv][2][c16] = sn2;
            if (pv) {
                Sout[Sb + 0*(size_t)P_ + px] = sn0;
                Sout[Sb + 1*(size_t)P_ + px] = sn1;
                Sout[Sb + 2*(size_t)P_ + px] = sn2;
                if (SpredOut) {
                    SpredOut[Sb + 0*(size_t)P_ + px] = sn0;
                    SpredOut[Sb + 1*(size_t)P_ + px] = sn1;
                    SpredOut[Sb + 2*(size_t)P_ + px] = sn2;
                }
            }
        }
        // C partials (lanes >=16 contribute zeros)
        q00 += sn0*sn0; q01 += sn0*sn1; q02 += sn0*sn2;
        q11 += sn1*sn1; q12 += sn1*sn2; q22 += sn2*sn2;

        // ---- phase 2: D(160m x 16n) += X_tile * Snew^T, K over 16 pixels ----
        v2f bf2[4];                              // B-frag: [k=p][col=n] = snew[n][p]
#pragma unroll
        for (int kc2 = 0; kc2 < 4; ++kc2) {
            float sbx = 0.f, sby = 0.f;
            if (c16 < 3) {
                sbx = snLds[wv][c16][kc2*4 + k0];
                sby = snLds[wv][c16][kc2*4 + k0 + 1];
            }
            bf2[kc2].x = sbx; bf2[kc2].y = sby;
        }
        // double-buffered A-fragments: loads for chunk mc+1 issue before the
        // 4 WMMAs of chunk mc, so the dscnt wait covers 4 loads at once.
        v2f afr[4], nxt[4];
#pragma unroll
        for (int kc2 = 0; kc2 < 4; ++kc2) {
            afr[kc2].x = xw[c16 * XSTRIDE + kc2*4 + k0];
            afr[kc2].y = xw[c16 * XSTRIDE + kc2*4 + k0 + 1];
        }
#pragma unroll
        for (int mc = 0; mc < 10; ++mc) {
            if (mc < 9) {
#pragma unroll
                for (int kc2 = 0; kc2 < 4; ++kc2) {
                    nxt[kc2].x = xw[((mc+1)*16 + c16) * XSTRIDE + kc2*4 + k0];
                    nxt[kc2].y = xw[((mc+1)*16 + c16) * XSTRIDE + kc2*4 + k0 + 1];
                }
            }
#pragma unroll
            for (int kc2 = 0; kc2 < 4; ++kc2) {
                dacc[mc] = __builtin_amdgcn_wmma_f32_16x16x4_f32(
                    false, afr[kc2], false, bf2[kc2], (short)0, dacc[mc],
                    false, false);
            }
#pragma unroll
            for (int kc2 = 0; kc2 < 4; ++kc2) afr[kc2] = nxt[kc2];
        }
    }

    // ---- write per-wave partials (no atomics) ----
#pragma unroll
    for (int off = 16; off >= 1; off >>= 1) {
        q00 += __shfl_xor(q00, off); q01 += __shfl_xor(q01, off);
        q02 += __shfl_xor(q02, off); q11 += __shfl_xor(q11, off);
        q12 += __shfl_xor(q12, off); q22 += __shfl_xor(q22, off);
    }
    if (lane == 0) {
        float* cp = Cpart + ((size_t)b * NWAVES + wave) * 8;
        cp[0]=q00; cp[1]=q01; cp[2]=q02; cp[3]=q11; cp[4]=q12; cp[5]=q22;
    }
    if (c16 < 3) {                               // lanes 0,1,2,16,17,18
        float* dp = Dpart + ((size_t)b * NWAVES + wave) * 480;
        const int rbase = (lane >> 4) * 8;
#pragma unroll
        for (int mc = 0; mc < 10; ++mc) {
#pragma unroll
            for (int i = 0; i < 8; ++i) {
                const int m = mc*16 + rbase + i;
                dp[m*3 + c16] = dacc[mc][i];
            }
        }
    }
}

// ---------------------------------------------------------------------------
// Kernel 2: reduce partials; A = proj(A - (A*C - D)/la); M3_next = W_{t+1}^T A
// ---------------------------------------------------------------------------
__device__ __forceinline__ int sym6(int a, int b2) {
    int lo = a < b2 ? a : b2, hi = a < b2 ? b2 : a;
    return lo == 0 ? hi : (lo == 1 ? 2 + hi : 5);
}

__global__ __launch_bounds__(256) void lpalm_a_update(
    const float* __restrict__ Aprev, const float* __restrict__ Wall,
    const float* __restrict__ L_A,
    const float* __restrict__ Cpart, const float* __restrict__ Dpart,
    float* __restrict__ AhistT, float* __restrict__ ApredOut,
    float* __restrict__ M3out, int t)
{
    __shared__ float c6s[6];
    __shared__ float aL[M_ * N_];
    __shared__ float normAcc[3];
    __shared__ float scl[3];

    const int b = blockIdx.x, tid = threadIdx.x;
    if (tid < 3) normAcc[tid] = 0.f;
    if (tid < 6) {
        float s = 0.f;
        for (int w = 0; w < NWAVES; ++w)
            s += Cpart[((size_t)b * NWAVES + w) * 8 + tid];
        c6s[tid] = s;
    }
    __syncthreads();

    const float la = L_A[t];
    const float* Ab = Aprev + (size_t)b * (M_ * N_);
    for (int j = tid; j < M_ * N_; j += 256) {
        float d = 0.f;
        for (int w = 0; w < NWAVES; ++w)
            d += Dpart[((size_t)b * NWAVES + w) * 480 + j];
        const int m = j / 3, n = j - 3 * m;
        const float a0 = Ab[m*3+0], a1 = Ab[m*3+1], a2 = Ab[m*3+2];
        const float ac = a0*c6s[sym6(0,n)] + a1*c6s[sym6(1,n)] + a2*c6s[sym6(2,n)];
        float anew = Ab[j] - (ac - d) / la;
        anew = fmaxf(anew, 0.f);
        aL[j] = anew;
        atomicAdd(&normAcc[n], anew * anew);
    }
    __syncthreads();
    if (tid < 3) {
        const float s = sqrtf(normAcc[tid]);
        scl[tid] = (s > 1.f) ? s : 1.f;
    }
    __syncthreads();
    for (int j = tid; j < M_ * N_; j += 256) {
        const int n = j % 3;
        const float af = aL[j] / scl[n];
        aL[j] = af;
        AhistT[(size_t)b * (M_ * N_) + j] = af;
        if (ApredOut) ApredOut[(size_t)b * (M_ * N_) + j] = af;
    }
    __syncthreads();
    if (t + 1 < T_ && tid < 9) {
        const int n = tid / 3, k = tid % 3;
        const float* Wn = Wall + (size_t)(t + 1) * (M_ * N_);
        float s = 0.f;
        for (int m = 0; m < M_; ++m) s += Wn[m*3 + n] * aL[m*3 + k];
        M3out[b * 16 + tid] = s;
    }
}

// M3_0 = W_0^T A_init
__global__ void lpalm_prep(const float* __restrict__ Ainit,
                           const float* __restrict__ Wall,
                           float* __restrict__ M3out)
{
    const int b = blockIdx.x, tid = threadIdx.x;
    if (tid < 9) {
        const int n = tid / 3, k = tid % 3;
        const float* Ab = Ainit + (size_t)b * (M_ * N_);
        float s = 0.f;
        for (int m = 0; m < M_; ++m) s += Wall[m*3 + n] * Ab[m*3 + k];
        M3out[b * 16 + tid] = s;
    }
}

extern "C" void kernel_launch(void* const* d_in, const int* in_sizes, int n_in,
                              void* d_out, int out_size, void* d_ws, size_t ws_size,
                              hipStream_t stream) {
    const float* X      = (const float*)d_in[0];
    const float* S_init = (const float*)d_in[1];
    const float* A_init = (const float*)d_in[2];
    const float* W      = (const float*)d_in[3];
    const float* theta  = (const float*)d_in[4];
    const float* L_A    = (const float*)d_in[5];

    float* out = (float*)d_out;
    const size_t SNP = (size_t)B_ * N_ * P_;   // 866400
    const size_t AMN = (size_t)B_ * M_ * N_;   // 14976
    float* S_pred = out;
    float* A_pred = out + SNP;
    float* S_hist = A_pred + AMN;
    float* A_hist = S_hist + (size_t)T_ * SNP;

    float* ws    = (float*)d_ws;
    float* M3ws  = ws;                         // 32*16
    float* Cpart = ws + 512;                   // 32*72*8
    float* Dpart = Cpart + (size_t)B_ * NWAVES * 8;  // 32*72*480

    lpalm_prep<<<dim3(B_), dim3(64), 0, stream>>>(A_init, W, M3ws);

    for (int t = 0; t < T_; ++t) {
        const float* Sprev = (t == 0) ? S_init : (S_hist + (size_t)(t - 1) * SNP);
        float* Sout = S_hist + (size_t)t * SNP;
        lpalm_s_update<<<dim3(NWAVES / WPB, B_), dim3(128), 0, stream>>>(
            X, Sprev, W, theta, M3ws, Sout,
            (t == T_ - 1) ? S_pred : nullptr, Cpart, Dpart, t);

        const float* Aprev = (t == 0) ? A_init : (A_hist + (size_t)(t - 1) * AMN);
        float* AhistT = A_hist + (size_t)t * AMN;
        lpalm_a_update<<<dim3(B_), dim3(256), 0, stream>>>(
            Aprev, W, L_A, Cpart, Dpart, AhistT,
            (t == T_ - 1) ? A_pred : nullptr, M3ws, t);
    }
}